// roi_pooling_fpn_creator_249108103706
// MI455X (gfx1250) — compile-verified
//
#include <hip/hip_runtime.h>

// ROI-Align (crop_and_resize 14x14 bilinear + 2x2 maxpool -> 7x7) for MI455X.
// Memory-bound gather kernel: feature map (62 MB) stays resident in the 192 MB
// L2; output (100 MB) is streamed out with non-temporal stores so it does not
// evict the feature map. No matrix structure -> no WMMA (see analysis).

typedef float v4f __attribute__((ext_vector_type(4)));

#define FH   200
#define FW   304
#define FC   256
#define CROP 14
#define POOL 7

struct Ax { int i0; int i1; float l; bool v; };

static __device__ __forceinline__ Ax make_axis(float s, int lim) {
    Ax a;
    float f = floorf(s);
    a.l = s - f;
    int i = (int)f;
    i = (i < 0) ? 0 : i;
    i = (i > lim) ? lim : i;
    a.i0 = i;
    int i1 = i + 1;
    a.i1 = (i1 > lim) ? lim : i1;
    a.v = (s >= 0.0f) && (s <= (float)lim);
    return a;
}

static __device__ __forceinline__ v4f vmax4(v4f a, v4f b) {
    v4f r;
    r.x = fmaxf(a.x, b.x);
    r.y = fmaxf(a.y, b.y);
    r.z = fmaxf(a.z, b.z);
    r.w = fmaxf(a.w, b.w);
    return r;
}

static __device__ __forceinline__ v4f sample_bilinear(const float* __restrict__ fbase,
                                                      const Ax& ay, const Ax& ax) {
    if (!(ay.v && ax.v)) {
        v4f z = {0.0f, 0.0f, 0.0f, 0.0f};
        return z;
    }
    const float* r0 = fbase + (size_t)ay.i0 * ((size_t)FW * FC);
    const float* r1 = fbase + (size_t)ay.i1 * ((size_t)FW * FC);
    v4f v00 = *(const v4f*)(r0 + (size_t)ax.i0 * FC);
    v4f v01 = *(const v4f*)(r0 + (size_t)ax.i1 * FC);
    v4f v10 = *(const v4f*)(r1 + (size_t)ax.i0 * FC);
    v4f v11 = *(const v4f*)(r1 + (size_t)ax.i1 * FC);
    v4f top = v00 + (v01 - v00) * ax.l;
    v4f bot = v10 + (v11 - v10) * ax.l;
    return top + (bot - top) * ay.l;
}

__global__ __launch_bounds__(256)
void roi_align_pool_kernel(const float* __restrict__ feat,       // [FH, FW, FC]
                           const float* __restrict__ proposals,  // [N, 4] x1,y1,x2,y2
                           const int*   __restrict__ image_shape,// [2] h,w
                           float*       __restrict__ out,        // [N, 7, 7, 256]
                           int nBoxes) {
    const int box = blockIdx.x;
    if (box >= nBoxes) return;

    const float h_img = (float)image_shape[0];
    const float w_img = (float)image_shape[1];

    const float px1 = proposals[(size_t)box * 4 + 0];
    const float py1 = proposals[(size_t)box * 4 + 1];
    const float px2 = proposals[(size_t)box * 4 + 2];
    const float py2 = proposals[(size_t)box * 4 + 3];

    // normalized box, then scale to feature-map coords exactly as reference
    const float y1n = py1 / h_img, y2n = py2 / h_img;
    const float x1n = px1 / w_img, x2n = px2 / w_img;

    const float ystart = y1n * (float)(FH - 1);
    const float ystep  = (y2n - y1n) * (float)(FH - 1) / (float)(CROP - 1);
    const float xstart = x1n * (float)(FW - 1);
    const float xstep  = (x2n - x1n) * (float)(FW - 1) / (float)(CROP - 1);

    const int t        = threadIdx.x;
    const int cg       = t & 63;   // channel quad: channels [4*cg, 4*cg+4)
    const int cellBase = t >> 6;   // 0..3

    const float* fbase = feat + (size_t)cg * 4;

    for (int cell = cellBase; cell < POOL * POOL; cell += 4) {
        const int pyi = cell / POOL;
        const int pxi = cell - pyi * POOL;

        const float ys0 = ystart + (float)(2 * pyi)     * ystep;
        const float ys1 = ystart + (float)(2 * pyi + 1) * ystep;
        const float xs0 = xstart + (float)(2 * pxi)     * xstep;
        const float xs1 = xstart + (float)(2 * pxi + 1) * xstep;

        const Ax ay0 = make_axis(ys0, FH - 1);
        const Ax ay1 = make_axis(ys1, FH - 1);
        const Ax ax0 = make_axis(xs0, FW - 1);
        const Ax ax1 = make_axis(xs1, FW - 1);

        v4f s00 = sample_bilinear(fbase, ay0, ax0);
        v4f s01 = sample_bilinear(fbase, ay0, ax1);
        v4f s10 = sample_bilinear(fbase, ay1, ax0);
        v4f s11 = sample_bilinear(fbase, ay1, ax1);

        v4f m = vmax4(vmax4(s00, s01), vmax4(s10, s11));

        // Streaming write-once output: non-temporal so the 100 MB of results
        // do not evict the L2-resident 62 MB feature map.
        float* dst = out + (((size_t)box * (POOL * POOL) + cell) * FC + (size_t)cg * 4);
        __builtin_nontemporal_store(m, (v4f*)dst);
    }
}

extern "C" void kernel_launch(void* const* d_in, const int* in_sizes, int n_in,
                              void* d_out, int out_size, void* d_ws, size_t ws_size,
                              hipStream_t stream) {
    const float* feat        = (const float*)d_in[0];   // [1, 200, 304, 256] fp32
    const float* proposals   = (const float*)d_in[1];   // [N, 4] fp32
    const int*   image_shape = (const int*)d_in[2];     // [2] int32
    float*       out         = (float*)d_out;           // [N, 7, 7, 256] fp32

    const int nBoxes = in_sizes[1] / 4;
    if (nBoxes <= 0) return;

    dim3 grid(nBoxes);
    dim3 block(256); // 8 wave32 waves; lanes of a wave share a cell -> 512B bursts
    roi_align_pool_kernel<<<grid, block, 0, stream>>>(feat, proposals, image_shape, out, nBoxes);
}